// HeteroGNN_5411658793574
// MI455X (gfx1250) — compile-verified
//
#include <hip/hip_runtime.h>

#define D 128
#define LDSROW 132   // 16x128 A-tile padded: lane stride 132 %64 = 4 banks -> conflict-free b64 reads

typedef float v2f __attribute__((ext_vector_type(2)));
typedef float v8f __attribute__((ext_vector_type(8)));

// ---------------------------------------------------------------------------
// K=128 accumulation into a 16x16 f32 tile via V_WMMA_F32_16X16X4_F32.
// WT=1: out[n] += sum_k a[k] * W[n*LDW + k]   ("x @ W^T", einsum 'ni,oi->no')
// WT=0: out[n] += sum_k a[k] * W[k*LDW + n]   ("x @ B",   einsum 'ni,io->no')
// LDW is compile-time so every B-load address is base + immediate offset.
// Two independent accumulators break the WMMA->WMMA RAW chain.
// ---------------------------------------------------------------------------
template <int WT, int LDW>
__device__ __forceinline__ void wmma_accum_k128(v8f& c0, v8f& c1, const float* As,
                                                const float* __restrict__ Wp,
                                                int n0, int lane) {
  const int l15 = lane & 15;
  const int hi  = lane >> 4;            // 0: K=kb+0/1   1: K=kb+2/3
  const int n   = n0 + l15;
  const float* ab = As + l15 * LDSROW + 2 * hi;
  const float* wb = WT ? (Wp + (long)n * LDW + 2 * hi)      // row n of W, pair select
                       : (Wp + (long)(2 * hi) * LDW + n);   // col n of B, pair select
#pragma unroll
  for (int kb = 0; kb < D; kb += 8) {
    v2f a0 = *(const v2f*)(ab + kb);
    v2f a1 = *(const v2f*)(ab + kb + 4);
    v2f b0, b1;
    if (WT) {
      b0 = *(const v2f*)(wb + kb);
      b1 = *(const v2f*)(wb + kb + 4);
    } else {
      b0.x = wb[kb * LDW];
      b0.y = wb[(kb + 1) * LDW];
      b1.x = wb[(kb + 4) * LDW];
      b1.y = wb[(kb + 5) * LDW];
    }
    c0 = __builtin_amdgcn_wmma_f32_16x16x4_f32(false, a0, false, b0,
                                               (short)0, c0, false, false);
    c1 = __builtin_amdgcn_wmma_f32_16x16x4_f32(false, a1, false, b1,
                                               (short)0, c1, false, false);
  }
}

// ---------------------------------------------------------------------------
// Fused dual GEMM: Out[m][:] = act( A1[g1(m)] x W1  (+ A2[g2(m)] x W2) + bias )
// scattered to Out[scatter(m)].  Block = 256 thr = 8 waves, covers 16 rows x 128 cols.
// blockIdx.y = attribute index a; per-a strides in elements.
// ---------------------------------------------------------------------------
template <int W1T, int LDW1, int W2T, int LDW2, bool HASA2>
__global__ __launch_bounds__(256) void gemm_dual_128(
    const float* __restrict__ A1, long a1S, int lda1,
    const float* __restrict__ W1, long w1S,
    const float* __restrict__ A2, long a2S, int lda2,
    const float* __restrict__ W2, long w2S,
    const float* __restrict__ bias, long biasS,
    float* __restrict__ Out, long outS,
    const int* __restrict__ gather1, const int* __restrict__ gather2,
    const int* __restrict__ scatter, int doRelu) {
  __shared__ float As[16 * LDSROW];
  const int tid  = threadIdx.x;
  const int lane = tid & 31;
  const int wave = tid >> 5;
  const int n0   = wave * 16;
  const int r0   = blockIdx.x * 16;
  const int a    = blockIdx.y;

  const float* A1p = A1 + (long)a * a1S;
  const float* W1p = W1 + (long)a * w1S;
  float*       Op  = Out + (long)a * outS;

  {
    const bool hasG = (gather1 != nullptr);
    for (int i = tid; i < 16 * D; i += 256) {     // stage A1 tile (coalesced)
      int r = i >> 7, cc = i & (D - 1);
      int gr = r0 + r;
      if (hasG) gr = gather1[gr];
      As[r * LDSROW + cc] = A1p[(long)gr * lda1 + cc];
    }
  }
  __syncthreads();

  v8f c0 = {}, c1 = {};
  wmma_accum_k128<W1T, LDW1>(c0, c1, As, W1p, n0, lane);

  if (HASA2) {
    const float* A2p = A2 + (long)a * a2S;
    const float* W2p = W2 + (long)a * w2S;
    __syncthreads();
    {
      const bool hasG = (gather2 != nullptr);
      for (int i = tid; i < 16 * D; i += 256) {
        int r = i >> 7, cc = i & (D - 1);
        int gr = r0 + r;
        if (hasG) gr = gather2[gr];
        As[r * LDSROW + cc] = A2p[(long)gr * lda2 + cc];
      }
    }
    __syncthreads();
    wmma_accum_k128<W2T, LDW2>(c0, c1, As, W2p, n0, lane);
  }

  // C layout: VGPR v, lanes 0-15 -> M=v; lanes 16-31 -> M=v+8; N = lane&15 (+n0)
  const int l15 = lane & 15;
  const int hi  = lane >> 4;
  const int col = n0 + l15;
  float b = 0.0f;
  if (bias) b = bias[(long)a * biasS + col];
  const bool hasS = (scatter != nullptr);
  const bool rl   = (doRelu != 0);
#pragma unroll
  for (int v = 0; v < 8; ++v) {
    int r  = r0 + (hi << 3) + v;
    int gr = hasS ? scatter[r] : r;
    float val = c0[v] + c1[v] + b;
    if (rl) val = fmaxf(val, 0.0f);
    Op[(long)gr * D + col] = val;
  }
}

// ---------------------------------------------------------------------------
// Segment-sum scatter: one wave per edge, float4 per lane (32 lanes * 4 = 128)
// sum[a][sidx[e]] += msgs[a][gidx[e]];  cnt[a][sidx[e]] += 1
// ---------------------------------------------------------------------------
__global__ __launch_bounds__(256) void seg_scatter_add(
    const float* __restrict__ msgs, long msgS,
    const int* __restrict__ gidx, const int* __restrict__ sidx, long idxS,
    float* __restrict__ sum, float* __restrict__ cnt, int nE, int nA, int Nn) {
  int idx  = blockIdx.x * 8 + (threadIdx.x >> 5);
  int lane = threadIdx.x & 31;
  if (idx >= nA * nE) return;
  int a = idx / nE;
  int e = idx - a * nE;
  int g = gidx[(long)a * idxS + e];
  int s = sidx[(long)a * idxS + e];
  float4 m = *(const float4*)(msgs + (long)a * msgS + (long)g * D + lane * 4);
  float* sp = sum + (long)a * Nn * D + (long)s * D + lane * 4;
  atomicAdd(sp + 0, m.x);
  atomicAdd(sp + 1, m.y);
  atomicAdd(sp + 2, m.z);
  atomicAdd(sp + 3, m.w);
  if (lane == 0) atomicAdd(cnt + (long)a * Nn + s, 1.0f);
}

__global__ __launch_bounds__(256) void seg_divide(float* __restrict__ sum,
                                                  const float* __restrict__ cnt,
                                                  long nvec4) {
  long i = (long)blockIdx.x * 256 + threadIdx.x;
  if (i >= nvec4) return;
  long node = i >> 5;                      // 32 float4 per 128-float row
  float c = fmaxf(cnt[node], 1.0f);
  float4* p = (float4*)sum + i;
  float4 v = *p;
  v.x /= c; v.y /= c; v.z /= c; v.w /= c;
  *p = v;
}

// ---------------------------------------------------------------------------
// Gauss-Jordan inverse of 128x128 proj[a]; Mout = inv(proj[a]) + I.
// One block per attribute; augmented [proj | I] lives in ws (128x256).
// ---------------------------------------------------------------------------
__global__ __launch_bounds__(256) void invert128(const float* __restrict__ proj,
                                                 float* __restrict__ aug,
                                                 float* __restrict__ Mout) {
  const int a   = blockIdx.x;
  const int tid = threadIdx.x;
  float* Ag = aug + (long)a * 128 * 256;
  for (int i = tid; i < 128 * 256; i += 256) {
    int r = i >> 8, cc = i & 255;
    Ag[i] = (cc < 128) ? proj[(long)a * 128 * 128 + r * 128 + cc]
                       : ((cc - 128 == r) ? 1.0f : 0.0f);
  }
  __shared__ float fct[128];
  __shared__ float pivrow[256];
  for (int p = 0; p < 128; ++p) {
    __syncthreads();
    float inv = 1.0f / Ag[p * 256 + p];
    pivrow[tid] = Ag[p * 256 + tid] * inv;      // scaled pivot row
    if (tid < 128) fct[tid] = Ag[tid * 256 + p];
    __syncthreads();
    Ag[p * 256 + tid] = pivrow[tid];
    for (int i = tid; i < 128 * 256; i += 256) {
      int r = i >> 8, cc = i & 255;
      if (r != p) Ag[i] -= fct[r] * pivrow[cc];
    }
  }
  __syncthreads();
  for (int i = tid; i < 128 * 128; i += 256) {
    int r = i >> 7, cc = i & 127;
    Mout[(long)a * 128 * 128 + i] = Ag[r * 256 + 128 + cc] + (r == cc ? 1.0f : 0.0f);
  }
}

// ---------------------------------------------------------------------------
extern "C" void kernel_launch(void* const* d_in, const int* in_sizes, int n_in,
                              void* d_out, int out_size, void* d_ws, size_t ws_size,
                              hipStream_t stream) {
  const int N = 100000, A = 3, P = 50000, E = 500000, EF = 500000;
  const float* x_ind = (const float*)d_in[0];
  const float* x_att = (const float*)d_in[1];
  const float* posf  = (const float*)d_in[2];
  const int*   pop   = (const int*)d_in[3];
  const int*   eia   = (const int*)d_in[4];   // (A,2,E): src = [a][0][:], dst = [a][1][:]
  const int*   fam   = (const int*)d_in[5];   // (2,EF)
  const float* proj  = (const float*)d_in[6];
  const float* aggW  = (const float*)d_in[7];
  const float* aggb  = (const float*)d_in[8];
  const float* Wl1   = (const float*)d_in[9];
  const float* bl1   = (const float*)d_in[10];
  const float* Wr1   = (const float*)d_in[11];
  const float* Wl2   = (const float*)d_in[12];
  const float* bl2   = (const float*)d_in[13];
  const float* Wr2   = (const float*)d_in[14];
  const float* Wl3   = (const float*)d_in[15];
  const float* bl3   = (const float*)d_in[16];
  const float* Wr3   = (const float*)d_in[17];

  const long AND = (long)A * N * D;
  float* ws  = (float*)d_ws;
  float* X0  = ws;                         // A*N*D  updated x_attributes
  float* X1  = X0 + AND;                   // A*N*D  layer-1 output
  float* G   = X1 + AND;                   // A*N*D  segment sums/means
  float* CNT = G + AND;                    // A*N    counts
  float* ESF = CNT + (long)A * N;          // A*P*D
  float* AUG = ESF + (long)A * P * D;      // A*128*256
  float* MIN = AUG + (long)A * 128 * 256;  // A*128*128  inv(proj)+I

  float* out_ind = (float*)d_out;          // N*D
  float* out_att = out_ind + (long)N * D;  // A*N*D

  // X0 <- x_attributes (population rows overwritten below)
  hipMemcpyAsync(X0, x_att, AND * sizeof(float), hipMemcpyDeviceToDevice, stream);

  // esf[a,p,:] = posf[p,a,:] @ proj[a]          (A@B orientation)
  gemm_dual_128<0, D, 0, D, false><<<dim3(P / 16, A), 256, 0, stream>>>(
      posf, (long)D, A * D, proj, (long)D * D,
      nullptr, 0, 0, nullptr, 0,
      nullptr, 0, ESF, (long)P * D, nullptr, nullptr, nullptr, 0);

  // X0[a, pop, :] = relu( x_att[a,pop] @ aggW[:, :128]^T + esf @ aggW[:,128:]^T + aggb )
  gemm_dual_128<1, 2 * D, 1, 2 * D, true><<<dim3(P / 16, A), 256, 0, stream>>>(
      x_att, (long)N * D, D, aggW, 0,
      ESF, (long)P * D, D, aggW + D, 0,
      aggb, 0, X0, (long)N * D, pop, nullptr, pop, 1);

  // agg1[a] = seg_mean( x_ind[src], dst )
  hipMemsetAsync(G, 0, AND * sizeof(float), stream);
  hipMemsetAsync(CNT, 0, (long)A * N * sizeof(float), stream);
  seg_scatter_add<<<(A * E + 7) / 8, 256, 0, stream>>>(
      x_ind, 0L, eia, eia + E, (long)2 * E, G, CNT, E, A, N);
  seg_divide<<<(unsigned)((AND / 4 + 255) / 256), 256, 0, stream>>>(G, CNT, AND / 4);

  // X1 = agg1 @ Wl1^T + bl1 + X0 @ Wr1^T
  gemm_dual_128<1, D, 1, D, true><<<dim3(N / 16, A), 256, 0, stream>>>(
      G, (long)N * D, D, Wl1, (long)D * D,
      X0, (long)N * D, D, Wr1, (long)D * D,
      bl1, (long)D, X1, (long)N * D, nullptr, nullptr, nullptr, 0);

  // M = inv(proj) + I
  invert128<<<A, 256, 0, stream>>>(proj, AUG, MIN);

  // X1[a, pop, :] = relu( X1[a, pop, :] @ M[a] )   (A@B, in-place row-parallel)
  gemm_dual_128<0, D, 0, D, false><<<dim3(P / 16, A), 256, 0, stream>>>(
      X1, (long)N * D, D, MIN, (long)D * D,
      nullptr, 0, 0, nullptr, 0,
      nullptr, 0, X1, (long)N * D, pop, nullptr, pop, 1);

  // agg2[a] = seg_mean( X1[dst], src )
  hipMemsetAsync(G, 0, AND * sizeof(float), stream);
  hipMemsetAsync(CNT, 0, (long)A * N * sizeof(float), stream);
  seg_scatter_add<<<(A * E + 7) / 8, 256, 0, stream>>>(
      X1, (long)N * D, eia + E, eia, (long)2 * E, G, CNT, E, A, N);
  seg_divide<<<(unsigned)((AND / 4 + 255) / 256), 256, 0, stream>>>(G, CNT, AND / 4);

  // out_att = agg2 @ Wl2^T + bl2 + X1 @ Wr2^T
  gemm_dual_128<1, D, 1, D, true><<<dim3(N / 16, A), 256, 0, stream>>>(
      G, (long)N * D, D, Wl2, (long)D * D,
      X1, (long)N * D, D, Wr2, (long)D * D,
      bl2, (long)D, out_att, (long)N * D, nullptr, nullptr, nullptr, 0);

  // agg3 = seg_mean( x_ind[fam[1]], fam[0] )
  hipMemsetAsync(G, 0, (long)N * D * sizeof(float), stream);
  hipMemsetAsync(CNT, 0, (long)N * sizeof(float), stream);
  seg_scatter_add<<<(EF + 7) / 8, 256, 0, stream>>>(
      x_ind, 0L, fam + EF, fam, 0L, G, CNT, EF, 1, N);
  seg_divide<<<(unsigned)(((long)N * D / 4 + 255) / 256), 256, 0, stream>>>(
      G, CNT, (long)N * D / 4);

  // out_ind = agg3 @ Wl3^T + bl3 + x_ind @ Wr3^T
  gemm_dual_128<1, D, 1, D, true><<<dim3(N / 16, 1), 256, 0, stream>>>(
      G, 0, D, Wl3, 0,
      x_ind, 0, D, Wr3, 0,
      bl3, 0, out_ind, 0, nullptr, nullptr, nullptr, 0);
}